// Decoder_35579509080225
// MI455X (gfx1250) — compile-verified
//
#include <hip/hip_runtime.h>
#include <stdint.h>

// ---------------- Problem constants (from the reference) ----------------
#define NUM_CLASSES 80
#define NUM_PRED    85            // 5 + NUM_CLASSES
#define THRESH      0.01f
#define B_DIM       8
#define HW_DIM      6400          // 80*80
#define A_DIM       3
#define ROWS_PER_B  (HW_DIM * A_DIM)     // 19200 anchor-rows per image (even)
#define NROWS       (B_DIM * ROWS_PER_B) // 153600 anchor-rows total
#define SEGS        8                     // class segments (blockIdx.y)
#define CLS_PER_SEG (NUM_CLASSES / SEGS)  // 10 classes per thread
#define BLOCK       256
#define ROWS_PER_BLOCK (2 * BLOCK)        // 512 rows per block (2 per thread)
#define MAX_OFF     176                   // >= ceil(512/3)+2 offset entries

typedef float v4f __attribute__((ext_vector_type(4)));

__device__ __forceinline__ float fast_sigmoid(float x) {
    // v_exp_f32 + v_add + v_rcp_f32; avoids the slow IEEE divide expansion.
    return __builtin_amdgcn_rcpf(1.0f + __expf(-x));
}

// One thread = TWO consecutive anchor rows (2r, 2r+1) x one 10-class segment.
//  - pairs never straddle a batch boundary (ROWS_PER_B is even)
//  - per class, a thread writes 48 contiguous bytes -> 3x 16B-aligned
//    global_store_b128 th:TH_NT; a wave's 32 lanes cover 1536B contiguous.
//  - block's grid-offset slice staged into LDS via global_load_async_to_lds_b64
//    (ASYNCcnt) so the 8 segments x 3 anchors reuse it from LDS, not global.
__global__ __launch_bounds__(BLOCK) void
decoder_kernel(const float* __restrict__ pred,    // (B, HW, A, 85) flat
               const float* __restrict__ anchor,  // (A, 2) flat
               const float* __restrict__ offset,  // (HW, 2) flat
               const int*   __restrict__ stride_p,
               float* __restrict__ out)           // (B, C, HW, A, 6) flat
{
    __shared__ float lds_off[MAX_OFF * 2];

    const int tid = threadIdx.x;
    const int R0  = blockIdx.x * ROWS_PER_BLOCK;

    // ---- async-stage this block's offset-table slice into LDS -------------
    // hw indices used by rows [R0, R0+512) form the contiguous (mod HW_DIM)
    // range hw_base .. hw_base+n_off-1, even across a batch wrap.
    const int i_first = R0 % ROWS_PER_B;
    const int hw_base = i_first / A_DIM;
    const int i_last  = (R0 + ROWS_PER_BLOCK - 1) % ROWS_PER_B;
    const int hw_last = i_last / A_DIM;
    int n_off = hw_last - hw_base;
    if (n_off < 0) n_off += HW_DIM;
    n_off += 1;                                   // <= 172 < MAX_OFF

    if (tid < n_off) {
        int hw = hw_base + tid;
        if (hw >= HW_DIM) hw -= HW_DIM;
        uint64_t gaddr = (uint64_t)(uintptr_t)(offset + (size_t)hw * 2);
        uint32_t laddr = (uint32_t)(uintptr_t)(&lds_off[tid * 2]);
        // async DMA: global (8B, aligned) -> LDS; tracked by ASYNCcnt
        asm volatile("global_load_async_to_lds_b64 %0, %1, off"
                     :: "v"(laddr), "v"(gaddr) : "memory");
    }
    asm volatile("s_wait_asynccnt 0x0" ::: "memory");  // drain this wave's DMA
    __syncthreads();                                   // publish LDS to block

    // ---- per-thread row pair ---------------------------------------------
    const int pair = blockIdx.x * BLOCK + tid;
    const int row0 = pair * 2;                   // even; row1 = row0+1 same b
    const int b    = row0 / ROWS_PER_B;
    const int i0   = row0 - b * ROWS_PER_B;      // even
    const int i1   = i0 + 1;
    const int ihw0 = i0 / A_DIM;
    const int a0   = i0 - ihw0 * A_DIM;
    const int ihw1 = i1 / A_DIM;
    const int a1   = i1 - ihw1 * A_DIM;

    const float* __restrict__ p0 = pred + (size_t)row0 * NUM_PRED;
    const float* __restrict__ p1 = p0 + NUM_PRED;
    const int c0 = blockIdx.y * CLS_PER_SEG;

    // warm the logit streams (global_prefetch_b8)
    __builtin_prefetch(p0 + 5 + c0, 0, 0);
    __builtin_prefetch(p1 + 5 + c0, 0, 0);

    // preamble: bbox + objectness for both rows (read-once NT stream)
    const float u0 = __builtin_nontemporal_load(p0 + 0);
    const float u1 = __builtin_nontemporal_load(p0 + 1);
    const float u2 = __builtin_nontemporal_load(p0 + 2);
    const float u3 = __builtin_nontemporal_load(p0 + 3);
    const float u4 = __builtin_nontemporal_load(p0 + 4);
    const float w0_ = __builtin_nontemporal_load(p1 + 0);
    const float w1_ = __builtin_nontemporal_load(p1 + 1);
    const float w2_ = __builtin_nontemporal_load(p1 + 2);
    const float w3_ = __builtin_nontemporal_load(p1 + 3);
    const float w4_ = __builtin_nontemporal_load(p1 + 4);

    int t0 = ihw0 - hw_base; if (t0 < 0) t0 += HW_DIM;
    int t1 = ihw1 - hw_base; if (t1 < 0) t1 += HW_DIM;
    const float ox0 = lds_off[t0 * 2 + 0], oy0 = lds_off[t0 * 2 + 1];
    const float ox1 = lds_off[t1 * 2 + 0], oy1 = lds_off[t1 * 2 + 1];

    const float st  = (float)stride_p[0];
    const float ax0 = anchor[a0 * 2 + 0], ay0 = anchor[a0 * 2 + 1];
    const float ax1 = anchor[a1 * 2 + 0], ay1 = anchor[a1 * 2 + 1];

    const float cx0 = (fast_sigmoid(u0) + ox0) * st;
    const float cy0 = (fast_sigmoid(u1) + oy0) * st;
    const float hx0 = 0.5f * __expf(u2) * ax0;
    const float hy0 = 0.5f * __expf(u3) * ay0;
    const float ob0 = fast_sigmoid(u4);
    const float x00 = cx0 - hx0, y00 = cy0 - hy0;
    const float x10 = cx0 + hx0, y10 = cy0 + hy0;

    const float cx1 = (fast_sigmoid(w0_) + ox1) * st;
    const float cy1 = (fast_sigmoid(w1_) + oy1) * st;
    const float hx1 = 0.5f * __expf(w2_) * ax1;
    const float hy1 = 0.5f * __expf(w3_) * ay1;
    const float ob1 = fast_sigmoid(w4_);
    const float x01 = cx1 - hx1, y01 = cy1 - hy1;
    const float x11 = cx1 + hx1, y11 = cy1 + hy1;

    // ---- stream the class segment ----------------------------------------
    // out rows for (b, c, i0..i1): (((b*C + c) * ROWS_PER_B) + i0) * 6 floats
    // byte offset is a multiple of 48 -> 16B-aligned b128 stores.
    float* __restrict__ o =
        out + (((size_t)(b * NUM_CLASSES + c0)) * ROWS_PER_B + (size_t)i0) * 6;
    const size_t class_stride = (size_t)ROWS_PER_B * 6;

    #pragma unroll
    for (int k = 0; k < CLS_PER_SEG; ++k) {
        const int   c  = c0 + k;
        const float l0 = __builtin_nontemporal_load(p0 + 5 + c);
        const float l1 = __builtin_nontemporal_load(p1 + 5 + c);
        const float s0 = fast_sigmoid(l0) * ob0;
        const float s1 = fast_sigmoid(l1) * ob1;
        const bool  m0 = s0 > THRESH;
        const bool  m1 = s1 > THRESH;
        const float cf = (float)c;
        const float cid0 = m0 ? cf : -1.0f, sc0 = m0 ? s0 : 0.0f;
        const float cid1 = m1 ? cf : -1.0f, sc1 = m1 ? s1 : 0.0f;

        v4f q0 = {cid0, sc0, x00, y00};     // row0: [cid, sc, x0, y0 | ...
        v4f q1 = {x10,  y10, cid1, sc1};    //        x1, y1] row1: [cid, sc |
        v4f q2 = {x01,  y01, x11,  y11};    //        x0, y0, x1, y1]
        __builtin_nontemporal_store(q0, (v4f*)(o + 0));
        __builtin_nontemporal_store(q1, (v4f*)(o + 4));
        __builtin_nontemporal_store(q2, (v4f*)(o + 8));
        o += class_stride;
    }
}

extern "C" void kernel_launch(void* const* d_in, const int* in_sizes, int n_in,
                              void* d_out, int out_size, void* d_ws, size_t ws_size,
                              hipStream_t stream) {
    const float* pred    = (const float*)d_in[0];  // (8,80,80,255) f32
    const float* anchor  = (const float*)d_in[1];  // (1,1,3,2)     f32
    const float* offset  = (const float*)d_in[2];  // (1,80,80,1,2) f32
    const int*   stridep = (const int*)d_in[3];    // scalar 8
    float*       out     = (float*)d_out;          // (8, 80*6400*3, 6) f32

    dim3 grid(NROWS / ROWS_PER_BLOCK, SEGS);       // (300, 8)
    decoder_kernel<<<grid, BLOCK, 0, stream>>>(pred, anchor, offset, stridep, out);
}